// TokenEmbedding_20005957664806
// MI455X (gfx1250) — compile-verified
//
#include <hip/hip_runtime.h>

// ---------------------------------------------------------------------------
// Dilated (tao=12, taps=8) circular width-3 conv cast as per-(b,c) GEMM:
//   Out(4096 x 24|32) = Feat(4096 x 24) * W(24 x 24|32)   via
//   V_WMMA_F32_16X16X4_F32, 6 k-steps of 4 over the 24 features f=m*3+k.
// Memory-bound (256MB store vs 3.2 GFLOP) -> fp32 WMMA; the strided channel
// gather x[b,:,c] (stride 84B) is done by the Tensor Data Mover as ONE
// tensor_load_to_lds DMA (2D descriptor, dim0_stride=21), not 128 vector
// gathers. Wave 0 issues + s_wait_tensorcnt, barrier publishes the LDS.
// ---------------------------------------------------------------------------

typedef float v2f __attribute__((ext_vector_type(2)));
typedef float v8f __attribute__((ext_vector_type(8)));
typedef unsigned int v4u __attribute__((ext_vector_type(4)));
typedef int          v4i __attribute__((ext_vector_type(4)));
typedef int          v8i __attribute__((ext_vector_type(8)));
typedef __attribute__((address_space(3))) float lds_float_t;

#define L_LEN  4096
#define C_CH   21
#define K_OUT  24
#define D_CH   512
#define TAO_   12
#define M_     7      // history reach = M*TAO = 84

// Issue one TDM load: copies every 21st float from gptr, 4096 in all, to consecutive floats at LDS offset lds_off  (16 KB gather DMA)
__device__ __forceinline__ void tdm_gather_trace(const float* gptr, unsigned lds_off)
{
    const unsigned long long ga = (unsigned long long)(uintptr_t)gptr;

    v4u g0;
    g0[0] = 1u;                                        // count=1, user desc
    g0[1] = lds_off;                                   // lds_addr (bytes)
    g0[2] = (unsigned)(ga & 0xFFFFFFFFu);              // global_addr[31:0]
    g0[3] = (unsigned)((ga >> 32) & 0x01FFFFFFu)       // global_addr[56:32]
          | (2u << 30);                                // type = 2 ("image")

    v8i g1;
    g1[0] = 0x00020000;                                // data_size=2 (4B), mask=0
    g1[1] = (int)(1u << 16);                           // tensor_dim0 = 1
    g1[2] = (int)((unsigned)L_LEN << 16);              // tensor_dim1 = 4096 (lo16)
    g1[3] = (int)(1u << 16);                           // tile_dim0 = 1
    g1[4] = L_LEN;                                     // tile_dim1 = 4096, tile_dim2=0
    g1[5] = C_CH;                                      // tensor_dim0_stride = 21 elems
    g1[6] = 0;                                         // (dim1_stride unused, 2D)
    g1[7] = 0;

    v4i gz = {0, 0, 0, 0};
#if defined(__clang_major__) && (__clang_major__ >= 23)
    v8i gz8 = {0, 0, 0, 0, 0, 0, 0, 0};
    __builtin_amdgcn_tensor_load_to_lds(g0, g1, gz, gz, gz8, 0);
#else
    __builtin_amdgcn_tensor_load_to_lds(g0, g1, gz, gz, 0);
#endif
}

__global__ __launch_bounds__(256)
void dilconv_wmma_f32(const float* __restrict__ x,
                      const float* __restrict__ conv_w,     // (24, 8, 3) -> [o*24 + f]
                      const float* __restrict__ conv_b,     // (24,)
                      const float* __restrict__ leftout_w,  // (8, 8, 3)  -> [o'*24 + f]
                      const float* __restrict__ leftout_b,  // (8,)
                      float* __restrict__ out)              // (32, 4096, 512)
{
    __shared__ float xs[L_LEN];                 // one channel trace, 16 KB

    const int blk  = blockIdx.x;                // 0 .. B*C-1
    const int b    = blk / C_CH;
    const int c    = blk - b * C_CH;
    const int tid  = threadIdx.x;
    const int wave = tid >> 5;                  // wave32: 8 waves / block
    const int lane = tid & 31;
    const int half = lane >> 4;                 // 0: lanes 0-15, 1: lanes 16-31
    const int ncol = lane & 15;
    const bool isLast = (c == C_CH - 1);        // c==20 also owns leftout cols

    // ---- TDM: DMA the strided channel trace x[b,:,c] into LDS (wave 0 only;
    //      TDM ignores EXEC, descriptor is uniform). Other waves overlap this
    //      with their weight loads; barrier publishes LDS.
    if (wave == 0) {
        const float* xb = x + (size_t)b * (L_LEN * C_CH) + c;
        const unsigned lds_off = (unsigned)(uintptr_t)(lds_float_t*)xs;
        tdm_gather_trace(xb, lds_off);
        __builtin_amdgcn_s_wait_tensorcnt(0);
    }

    // ---- per-lane B fragments for all 6 k-steps (loop-invariant over t)
    // B 4x16 f32 layout: lane<16 -> v0=K(4kk+0), v1=K(4kk+1); lane>=16 -> K+2,K+3
    v2f bw0[6], bw1[6];
#pragma unroll
    for (int kk = 0; kk < 6; ++kk) {
        const int f0 = kk * 4 + half * 2;
        bw0[kk][0] = conv_w[ncol * 24 + f0];
        bw0[kk][1] = conv_w[ncol * 24 + f0 + 1];
        const int c1 = 16 + ncol;               // N-tile 1 column (16..31)
        float wa = 0.0f, wb = 0.0f;
        if (c1 < K_OUT) {                       // still conv outputs (16..23)
            wa = conv_w[c1 * 24 + f0];
            wb = conv_w[c1 * 24 + f0 + 1];
        } else if (isLast) {                    // leftout outputs (24..31)
            wa = leftout_w[(c1 - K_OUT) * 24 + f0];
            wb = leftout_w[(c1 - K_OUT) * 24 + f0 + 1];
        }
        bw1[kk][0] = wa; bw1[kk][1] = wb;
    }
    const float bias0 = conv_b[ncol];
    const int   cc1   = 16 + ncol;
    const float bias1 = (cc1 < K_OUT) ? conv_b[cc1]
                      : (isLast ? leftout_b[cc1 - K_OUT] : 0.0f);

    __syncthreads();                            // LDS trace now visible to all

    float* outb = out + (size_t)b * L_LEN * D_CH + (size_t)c * K_OUT;

    // ---- each wave: 32 tiles of 16 rows x 32 output channels
    for (int tile = wave; tile < L_LEN / 16; tile += 8) {
        const int t0   = tile * 16;
        const int trow = t0 + ncol;             // A row for this lane

        v8f acc0, acc1;                         // C init = bias (same col per VGPR)
#pragma unroll
        for (int v = 0; v < 8; ++v) { acc0[v] = bias0; acc1[v] = bias1; }

#pragma unroll
        for (int kk = 0; kk < 6; ++kk) {
            const int f0 = kk * 4 + half * 2;   // features this lane supplies
            v2f a;
#pragma unroll
            for (int e = 0; e < 2; ++e) {
                const int f  = f0 + e;
                const int m  = f / 3;
                const int k  = f - 3 * m;
                const int tp = (trow + k - 1) & (L_LEN - 1);   // circular pad
                int ts = tp - TAO_ * m;                        // >=0 when tp>=84
                if (ts < 0) ts = 0;
                const float xv = xs[ts];
                a[e] = (tp >= M_ * TAO_) ? xv : 0.0f;          // left zero-pad
            }
            acc0 = __builtin_amdgcn_wmma_f32_16x16x4_f32(
                       false, a, false, bw0[kk], (short)0, acc0, false, false);
            acc1 = __builtin_amdgcn_wmma_f32_16x16x4_f32(
                       false, a, false, bw1[kk], (short)0, acc1, false, false);
        }

        // ---- store: D[v][lane] = row v+8*half, col lane&15 -> 64B row segments
        const bool storeHi = (ncol < 8) || isLast;  // c<20 has only 24 valid cols
#pragma unroll
        for (int v = 0; v < 8; ++v) {
            const int r = v + half * 8;
            float* p = outb + (size_t)(t0 + r) * D_CH;
            p[ncol] = acc0[v];
            if (storeHi) p[16 + ncol] = acc1[v];
        }
    }
}

extern "C" void kernel_launch(void* const* d_in, const int* in_sizes, int n_in,
                              void* d_out, int out_size, void* d_ws, size_t ws_size,
                              hipStream_t stream) {
    const float* x         = (const float*)d_in[0];
    const float* conv_w    = (const float*)d_in[1];
    const float* conv_b    = (const float*)d_in[2];
    const float* leftout_w = (const float*)d_in[3];
    const float* leftout_b = (const float*)d_in[4];
    float* out = (float*)d_out;

    dim3 grid(32 * C_CH);   // one block per (batch, channel)
    dim3 block(256);        // 8 wave32 waves
    dilconv_wmma_f32<<<grid, block, 0, stream>>>(x, conv_w, conv_b,
                                                 leftout_w, leftout_b, out);
}